// LGConv_936302871075
// MI455X (gfx1250) — compile-verified
//
#include <hip/hip_runtime.h>

#define WPB 8  // waves per block: 256 threads / wave32

// ---------------- zero out + deg ----------------
__global__ void lgconv_zero_kernel(float4* __restrict__ out4, long long n4,
                                   float* __restrict__ deg, int n) {
  long long i  = (long long)blockIdx.x * blockDim.x + threadIdx.x;
  long long gs = (long long)gridDim.x * blockDim.x;
  for (long long k = i; k < n4; k += gs) out4[k] = make_float4(0.f, 0.f, 0.f, 0.f);
  for (long long k = i; k < (long long)n; k += gs) deg[k] = 0.f;
}

// ---------------- in-degree via hw f32 atomics ----------------
__global__ void lgconv_degree_kernel(const int* __restrict__ dst,
                                     float* __restrict__ deg, int E) {
  long long i  = (long long)blockIdx.x * blockDim.x + threadIdx.x;
  long long gs = (long long)gridDim.x * blockDim.x;
  for (long long k = i; k < (long long)E; k += gs)
    unsafeAtomicAdd(&deg[dst[k]], 1.0f);
}

// ---------------- deg -> deg^{-1/2} in place ----------------
__global__ void lgconv_rsqrt_kernel(float* __restrict__ deg, int n) {
  int i = blockIdx.x * blockDim.x + threadIdx.x;
  if (i < n) {
    float d = deg[i];
    deg[i] = (d > 0.f) ? rsqrtf(d) : 0.f;  // 0 for zero in-degree, else rsqrt(d)
  }
}

// ---------------- gather-scale-scatter, wave per edge, async double-buffer ----------------
__global__ __launch_bounds__(256) void lgconv_scatter_kernel(
    const float* __restrict__ x, const int* __restrict__ src,
    const int* __restrict__ dst, const float* __restrict__ dis,
    float* __restrict__ out, int E) {
  __shared__ float smem[WPB * 128];  // 2 x 64-float row buffers per wave

  const int lane = threadIdx.x & 31;
  const int wv   = threadIdx.x >> 5;
  const long long wid    = (long long)blockIdx.x * WPB + wv;
  const long long stride = (long long)gridDim.x * WPB;

  float* buf0 = &smem[wv * 128];
  float* buf1 = buf0 + 64;
  // low 32 bits of the flat shared pointer == DS byte offset
  const unsigned lds0 = (unsigned)(size_t)buf0 + (unsigned)(lane << 3);
  const unsigned lds1 = lds0 + 256u;

  long long e = wid;
  if (e >= (long long)E) return;  // wave-uniform exit

  // prologue: kick off async gather of row x[src[e]] -> buf0 (32 lanes x 8B)
  int s_cur = src[e];
  {
    const float* g = x + ((long long)s_cur << 6) + (lane << 1);
    asm volatile("global_load_async_to_lds_b64 %0, %1, off"
                 :: "v"(lds0), "v"(g) : "memory");
  }

  unsigned cur = 0;
  for (; e < (long long)E; e += stride) {
    const long long en = e + stride;
    int s_next = 0;
    if (en < (long long)E) {
      s_next = src[en];
      const float*   g    = x + ((long long)s_next << 6) + (lane << 1);
      const unsigned ldsn = cur ? lds0 : lds1;
      asm volatile("global_load_async_to_lds_b64 %0, %1, off"
                   :: "v"(ldsn), "v"(g) : "memory");
      // 2 outstanding, in-order retire: <=1 left means current buffer is done
      asm volatile("s_wait_asynccnt 0x1" ::: "memory");
    } else {
      asm volatile("s_wait_asynccnt 0x0" ::: "memory");
    }

    const int   d    = dst[e];
    const float norm = dis[s_cur] * dis[d];

    const float* cb = cur ? buf1 : buf0;
    const float2 v  = *(const float2*)(cb + (lane << 1));  // ds_load_b64

    float* o = out + ((long long)d << 6) + (lane << 1);
    unsafeAtomicAdd(o,     v.x * norm);   // global_atomic_add_f32
    unsafeAtomicAdd(o + 1, v.y * norm);

    s_cur = s_next;
    cur ^= 1u;
  }
}

extern "C" void kernel_launch(void* const* d_in, const int* in_sizes, int n_in,
                              void* d_out, int out_size, void* d_ws, size_t ws_size,
                              hipStream_t stream) {
  const float* x    = (const float*)d_in[0];
  const int*   edge = (const int*)d_in[1];
  const int N = in_sizes[0] / 64;
  const int E = in_sizes[1] / 2;
  const int* src = edge;       // edge_index[0,:]
  const int* dst = edge + E;   // edge_index[1,:]
  float* out = (float*)d_out;
  float* deg = (float*)d_ws;   // N floats of scratch

  {  // zero out (as float4) and deg
    long long n4 = (long long)N * 16;
    int blocks = (int)((n4 + 255) / 256);
    if (blocks > 2048) blocks = 2048;
    if (blocks < 1) blocks = 1;
    lgconv_zero_kernel<<<blocks, 256, 0, stream>>>((float4*)out, n4, deg, N);
  }
  {  // in-degree
    int blocks = (E + 255) / 256;
    if (blocks > 2048) blocks = 2048;
    if (blocks < 1) blocks = 1;
    lgconv_degree_kernel<<<blocks, 256, 0, stream>>>(dst, deg, E);
  }
  lgconv_rsqrt_kernel<<<(N + 255) / 256, 256, 0, stream>>>(deg, N);
  {  // main gather-scale-scatter
    int blocks = (int)(((long long)E + WPB - 1) / WPB);
    if (blocks > 4096) blocks = 4096;
    if (blocks < 1) blocks = 1;
    lgconv_scatter_kernel<<<blocks, 256, 0, stream>>>(x, src, dst, deg, out, E);
  }
}